// TensorProduct_65901978189853
// MI455X (gfx1250) — compile-verified
//
#include <hip/hip_runtime.h>
#include <stdint.h>

#define TPB     256          // 8 wave32s; 128 threads per row, 2 rows per iter
#define ROW_F   512          // floats per x1 row
#define OUT_F   896          // floats per output row
#define CHUNK_B 4096         // bytes per 2-row x1 chunk (TPB lanes * 16B)

// Async DMA one 16B slice of a 2-row x1 chunk into LDS (ASYNCcnt-tracked).
// Always issues (clamped address) so per-wave ASYNCcnt bookkeeping stays exact.
__device__ __forceinline__ void async_copy_16B(const float* __restrict__ x1,
                                               size_t x1Bytes,
                                               uint32_t ldsByte,
                                               size_t gByte)
{
    if (gByte + 16 > x1Bytes) gByte = x1Bytes - 16;   // clamp, never skip
    const char* g = (const char*)x1 + gByte;
    asm volatile("global_load_async_to_lds_b128 %0, %1, off"
                 :: "v"(ldsByte), "v"(g)
                 : "memory");
}

__global__ __launch_bounds__(TPB)
void tp_uvu_kernel(const float* __restrict__ x1,
                   const float* __restrict__ x2,
                   const float* __restrict__ w,
                   float* __restrict__ out,
                   int N, int pairsPerBlock, int totalPairs)
{
    __shared__ float smem[2 * 2 * ROW_F];   // double buffer x (2 rows) = 8 KB

    const int tid = threadIdx.x;            // 0..255
    const int u   = tid & 127;              // irrep channel
    const int ty  = tid >> 7;               // row within 2-row chunk

    // Fold CG norms + path alphas into per-u weights (registers, once per block).
    //   A0 = sqrt(1/2); A1/sqrt3 = sqrt(1/2); A0/sqrt3 = sqrt(1/6); A2/sqrt6 = sqrt(1/2)
    const float S_HALF  = 0.70710678118654752440f;  // sqrt(1/2)
    const float S_SIXTH = 0.40824829046386301636f;  // sqrt(1/6)
    const float W0 = S_HALF  * w[u];          // 0e x 0e -> 0e
    const float W1 = S_HALF  * w[128 + u];    // 0e x 1o -> 1o
    const float W2 = S_HALF  * w[256 + u];    // 1o x 0e -> 1o
    const float W3 = S_SIXTH * w[384 + u];    // 1o . 1o -> 0e
    const float W4 = S_HALF  * w[512 + u];    // 1o x 1o -> 1e (cross)

    const size_t   x1Bytes = (size_t)N * (ROW_F * 4);
    const uint32_t ldsBase = (uint32_t)(uintptr_t)(&smem[0]) + (uint32_t)(tid * 16);

    int pStart = blockIdx.x * pairsPerBlock;
    int pEnd   = pStart + pairsPerBlock;
    if (pEnd > totalPairs) pEnd = totalPairs;
    const int iters = pEnd - pStart;
    if (iters <= 0) return;

    // --- pipeline prologue: prefetch chunks 0 and 1 ---
    async_copy_16B(x1, x1Bytes, ldsBase,
                   (size_t)pStart * CHUNK_B + (size_t)tid * 16);
    if (iters > 1)
        async_copy_16B(x1, x1Bytes, ldsBase + CHUNK_B,
                       ((size_t)pStart + 1) * CHUNK_B + (size_t)tid * 16);

    for (int it = 0; it < iters; ++it) {
        const int buf = it & 1;

        // Drain down to the still-in-flight next chunk (loads complete in order).
        if (it + 1 < iters)
            asm volatile("s_wait_asynccnt 0x1" ::: "memory");
        else
            asm volatile("s_wait_asynccnt 0x0" ::: "memory");
        __syncthreads();   // make async-written LDS visible across all 8 waves

        const int n = (pStart + it) * 2 + ty;
        if (n < N) {
            const float* row = &smem[buf * (2 * ROW_F) + ty * ROW_F];
            const float s  = row[u];
            const float v0 = row[128 + 3 * u + 0];
            const float v1 = row[128 + 3 * u + 1];
            const float v2 = row[128 + 3 * u + 2];

            // x2 row: wave-uniform 16B load, L2 resident (1.6 MB total)
            const float4 q = ((const float4* __restrict__)x2)[n];
            const float a = q.x, b0 = q.y, b1 = q.z, b2 = q.w;

            const float o0 = W0 * (s * a) + W3 * (v0 * b0 + v1 * b1 + v2 * b2);

            const float sw = W1 * s;
            const float aw = W2 * a;
            const float p0 = sw * b0 + aw * v0;
            const float p1 = sw * b1 + aw * v1;
            const float p2 = sw * b2 + aw * v2;

            const float c0 = W4 * (v1 * b2 - v2 * b1);
            const float c1 = W4 * (v2 * b0 - v0 * b2);
            const float c2 = W4 * (v0 * b1 - v1 * b0);

            float* __restrict__ orow = out + (size_t)n * OUT_F;
            orow[u] = o0;                       // 128x0e  (b32, coalesced)
            orow[128 + 3 * u + 0] = p0;         // 128x1o  (12B/lane, contiguous)
            orow[128 + 3 * u + 1] = p1;
            orow[128 + 3 * u + 2] = p2;
            orow[512 + 3 * u + 0] = c0;         // 128x1e  (12B/lane, contiguous)
            orow[512 + 3 * u + 1] = c1;
            orow[512 + 3 * u + 2] = c2;
        }

        __syncthreads();   // all waves done reading `buf` before DMA overwrites it
        if (it + 2 < iters)
            async_copy_16B(x1, x1Bytes, ldsBase + buf * CHUNK_B,
                           ((size_t)(pStart + it) + 2) * CHUNK_B + (size_t)tid * 16);
    }
}

extern "C" void kernel_launch(void* const* d_in, const int* in_sizes, int n_in,
                              void* d_out, int out_size, void* d_ws, size_t ws_size,
                              hipStream_t stream)
{
    const float* x1 = (const float*)d_in[0];   // (N, 512) fp32
    const float* x2 = (const float*)d_in[1];   // (N, 4)   fp32
    const float* w  = (const float*)d_in[2];   // (640,)   fp32
    float* out      = (float*)d_out;           // (N, 896) fp32

    const int N          = in_sizes[0] / (4 * 128);
    const int totalPairs = (N + 1) / 2;

    int grid = 2048;
    if (grid > totalPairs) grid = totalPairs;
    const int ppb = (totalPairs + grid - 1) / grid;

    tp_uvu_kernel<<<dim3(grid), dim3(TPB), 0, stream>>>(x1, x2, w, out,
                                                        N, ppb, totalPairs);
}